// NGMConvLayer_28913719836845
// MI455X (gfx1250) — compile-verified
//
#include <hip/hip_runtime.h>
#include <hip/hip_bf16.h>

typedef float v2f __attribute__((ext_vector_type(2)));
typedef float v8f __attribute__((ext_vector_type(8)));
typedef float f4  __attribute__((ext_vector_type(4)));

#define B_ 8
#define N_ 256
#define F_ 64
#define ROWS_ (B_ * N_)   // 2048

// ---------------------------------------------------------------------------
// MLP kernel: out[r,:] = relu(relu(x[r,:] @ w1 + b1) @ w2 + b2)
// rows = 2048, F = 64.  One wave (32 lanes) computes a 16-row x 64-col tile
// with V_WMMA_F32_16X16X4_F32; K=64 split into 16 chunks of 4.
// Fragment layouts per CDNA5 ISA 7.12.2:
//   A 16x4 f32 : lane m = lane&15 holds row m; VGPR v holds K = 2*(lane>>4)+v
//   B 4x16 f32 : lane n = lane&15 holds col n; VGPR v holds K = 2*(lane>>4)+v
//   C/D 16x16  : VGPR r, lane -> (M = r + 8*(lane>>4), N = lane&15)
// ---------------------------------------------------------------------------
__global__ __launch_bounds__(256) void mlp_wmma_kernel(
    const float* __restrict__ x,
    const float* __restrict__ w1, const float* __restrict__ b1,
    const float* __restrict__ w2, const float* __restrict__ b2,
    float* __restrict__ out)
{
    __shared__ float hbuf[8][16 * 68];   // per-wave 16x64 tile, stride 68 (pad)

    const int lane = threadIdx.x & 31;
    const int wv   = threadIdx.x >> 5;
    const int m    = lane & 15;
    const int hi   = lane >> 4;
    const int row0 = (blockIdx.x * 8 + wv) * 16;

    // ---------------- layer 1 ----------------
    v2f a[16];
    const float* xr = x + (size_t)(row0 + m) * F_;
#pragma unroll
    for (int kc = 0; kc < 16; ++kc) {
        const int k = 4 * kc + 2 * hi;
        a[kc][0] = xr[k];
        a[kc][1] = xr[k + 1];
    }

    v8f c[4] = {};
#pragma unroll
    for (int n = 0; n < 4; ++n) {
#pragma unroll
        for (int kc = 0; kc < 16; ++kc) {
            const int k = 4 * kc + 2 * hi;
            v2f bf;
            bf[0] = w1[(size_t)(k + 0) * F_ + n * 16 + m];
            bf[1] = w1[(size_t)(k + 1) * F_ + n * 16 + m];
            c[n] = __builtin_amdgcn_wmma_f32_16x16x4_f32(
                false, a[kc], false, bf, (short)0, c[n], false, false);
        }
    }

    // bias + relu -> LDS (h tile, row-major, padded stride 68)
    float* hl = hbuf[wv];
#pragma unroll
    for (int n = 0; n < 4; ++n) {
        const int   ncol = n * 16 + m;
        const float bias = b1[ncol];
#pragma unroll
        for (int r = 0; r < 8; ++r) {
            const int mr = r + 8 * hi;
            hl[mr * 68 + ncol] = fmaxf(c[n][r] + bias, 0.0f);
        }
    }
    __syncthreads();

    // ---------------- layer 2 ----------------
    v2f a2[16];
#pragma unroll
    for (int kc = 0; kc < 16; ++kc) {
        const int k = 4 * kc + 2 * hi;
        a2[kc][0] = hl[m * 68 + k];
        a2[kc][1] = hl[m * 68 + k + 1];
    }

    v8f d[4] = {};
#pragma unroll
    for (int n = 0; n < 4; ++n) {
#pragma unroll
        for (int kc = 0; kc < 16; ++kc) {
            const int k = 4 * kc + 2 * hi;
            v2f bf;
            bf[0] = w2[(size_t)(k + 0) * F_ + n * 16 + m];
            bf[1] = w2[(size_t)(k + 1) * F_ + n * 16 + m];
            d[n] = __builtin_amdgcn_wmma_f32_16x16x4_f32(
                false, a2[kc], false, bf, (short)0, d[n], false, false);
        }
    }

#pragma unroll
    for (int n = 0; n < 4; ++n) {
        const int   ncol = n * 16 + m;
        const float bias = b2[ncol];
#pragma unroll
        for (int r = 0; r < 8; ++r) {
            const int mr = r + 8 * hi;
            out[(size_t)(row0 + mr) * F_ + ncol] = fmaxf(d[n][r] + bias, 0.0f);
        }
    }
}

// ---------------------------------------------------------------------------
// Fused kernel: one block per (b,i) row.
//   1) L1-normalize A[b,i,:] (LDS reduction)
//   2) stream W[b,i,:,:] (16384 floats) ONCE with non-temporal float4 loads,
//      copy it to out (non-temporal stores), accumulate
//      x2[b,i,e] = sum_j An[j] * W[j,e] * x1[b,j,e] on the fly
//   3) cross-group reduce, add self-MLP term, write x2
// Thread t owns e-quad q = t&15 and j = t/16 + 16k (k=0..15).
// ---------------------------------------------------------------------------
__global__ __launch_bounds__(256) void fuse_norm_einsum_copy_kernel(
    const float* __restrict__ A,
    const float* __restrict__ W,
    const float* __restrict__ x1,
    const float* __restrict__ xs,
    float* __restrict__ outW,
    float* __restrict__ outX2)
{
    __shared__ float red[256];
    __shared__ float An[256];
    __shared__ f4    part[256];

    const int t   = threadIdx.x;
    const int row = blockIdx.x;          // row = b*256 + i
    const int b   = row >> 8;

    // ---- L1 normalization of adjacency row ----
    const float aval = A[(size_t)row * N_ + t];
    red[t] = fabsf(aval);
    __syncthreads();
#pragma unroll
    for (int s = 128; s > 0; s >>= 1) {
        if (t < s) red[t] += red[t + s];
        __syncthreads();
    }
    const float denom = fmaxf(red[0], 1e-12f);
    An[t] = aval / denom;
    __syncthreads();

    // ---- stream W row: copy + weighted accumulate ----
    const f4* __restrict__ Wv = (const f4*)(W    + (size_t)row * (N_ * F_));
    f4* __restrict__       Ov = (f4*)      (outW + (size_t)row * (N_ * F_));
    const f4* __restrict__ Xv = (const f4*)(x1   + (size_t)b   * (N_ * F_));

    const int q = t & 15;
    const int g = t >> 4;
    f4 acc = {0.0f, 0.0f, 0.0f, 0.0f};
#pragma unroll
    for (int k = 0; k < 16; ++k) {
        const int c = t + 256 * k;
        const int j = g + 16 * k;
        f4 w = __builtin_nontemporal_load(&Wv[c]);
        if (k < 15) __builtin_prefetch(&Wv[c + 256], 0, 0);
        __builtin_nontemporal_store(w, &Ov[c]);
        const float an = An[j];
        const f4    xv = Xv[j * 16 + q];
        acc.x = fmaf(an * w.x, xv.x, acc.x);
        acc.y = fmaf(an * w.y, xv.y, acc.y);
        acc.z = fmaf(an * w.z, xv.z, acc.z);
        acc.w = fmaf(an * w.w, xv.w, acc.w);
    }

    // ---- reduce 16 j-groups per e-quad, add self term, write x2 ----
    part[t] = acc;
    __syncthreads();
    if (t < 16) {
        f4 s = part[t];
#pragma unroll
        for (int g2 = 1; g2 < 16; ++g2) {
            const f4 p = part[t + 16 * g2];
            s.x += p.x; s.y += p.y; s.z += p.z; s.w += p.w;
        }
        const f4 sv = ((const f4*)(xs + (size_t)row * F_))[t];
        s.x += sv.x; s.y += sv.y; s.z += sv.z; s.w += sv.w;
        ((f4*)(outX2 + (size_t)row * F_))[t] = s;
    }
}

extern "C" void kernel_launch(void* const* d_in, const int* in_sizes, int n_in,
                              void* d_out, int out_size, void* d_ws, size_t ws_size,
                              hipStream_t stream) {
    const float* A    = (const float*)d_in[0];
    const float* W    = (const float*)d_in[1];
    const float* x    = (const float*)d_in[2];
    const float* n_w1 = (const float*)d_in[3];
    const float* n_b1 = (const float*)d_in[4];
    const float* n_w2 = (const float*)d_in[5];
    const float* n_b2 = (const float*)d_in[6];
    const float* s_w1 = (const float*)d_in[7];
    const float* s_b1 = (const float*)d_in[8];
    const float* s_w2 = (const float*)d_in[9];
    const float* s_b2 = (const float*)d_in[10];

    float* x1 = (float*)d_ws;                         // [2048,64]
    float* xs = x1 + (size_t)ROWS_ * F_;              // [2048,64]

    float* outW  = (float*)d_out;                     // W passthrough (33.5M f32)
    float* outX2 = outW + (size_t)B_ * N_ * N_ * F_;  // x2 (131072 f32)

    // 128 row-tiles of 16 rows, 8 waves per block -> 16 blocks
    mlp_wmma_kernel<<<16, 256, 0, stream>>>(x, n_w1, n_b1, n_w2, n_b2, x1);
    mlp_wmma_kernel<<<16, 256, 0, stream>>>(x, s_w1, s_b1, s_w2, s_b2, xs);
    fuse_norm_einsum_copy_kernel<<<ROWS_, 256, 0, stream>>>(A, W, x1, xs, outW, outX2);
}